// GraphClassifier_80659485819336
// MI455X (gfx1250) — compile-verified
//
#include <hip/hip_runtime.h>

#define HID   128
#define NGR   512
#define NCLS  10

typedef float v2f __attribute__((ext_vector_type(2)));
typedef float v8f __attribute__((ext_vector_type(8)));

// ---------------- elementwise / setup kernels ----------------

__global__ void k_fill(float* __restrict__ p, int n, float v) {
  int i = blockIdx.x * blockDim.x + threadIdx.x;
  if (i < n) p[i] = v;
}

__global__ void k_degree(const int* __restrict__ dst, float* __restrict__ deg, int E) {
  int e = blockIdx.x * blockDim.x + threadIdx.x;
  if (e < E) atomicAdd(&deg[dst[e]], 1.0f);
}

__global__ void k_dinv(const float* __restrict__ deg, float* __restrict__ dinv, int N) {
  int i = blockIdx.x * blockDim.x + threadIdx.x;
  if (i < N) dinv[i] = rsqrtf(deg[i] + 1.0f);
}

// ---------------- main fp32 WMMA GEMM, K=N=128, strip form ----------------
// One wave computes a 16x128 output strip (8 accumulators). B (128x128, 64KB) is
// staged once per block into LDS with an XOR-16 column swizzle keyed on row-pair
// parity so the two lane-halves of each ds_load hit disjoint 16-bank windows.
// Epilogue writes C = A@B and C2 = C * dinv[row]^2 (GCN self-loop term fused).
__global__ void gemm_wmma_strip_k128(const float* __restrict__ A,
                                     const float* __restrict__ B,
                                     float* __restrict__ C,
                                     float* __restrict__ C2,
                                     const float* __restrict__ dinv, int M) {
  __shared__ float Bs[128 * 128];

  // cooperative swizzled staging of B (all threads, before any wave exits)
  for (int idx = threadIdx.x; idx < 128 * 128; idx += blockDim.x) {
    const int row = idx >> 7;
    const int col = idx & 127;
    Bs[(row << 7) | (col ^ (((row >> 1) & 1) << 4))] = B[idx];
  }
  __syncthreads();

  const int lane = threadIdx.x & 31;
  const int wave = threadIdx.x >> 5;
  const int tm   = blockIdx.x * (blockDim.x >> 5) + wave;   // wave-uniform
  if (tm >= (M >> 4)) return;                               // whole-wave exit

  const int half = lane >> 4;       // 0 | 1
  const int l15  = lane & 15;
  const int swz  = half << 4;       // rows {k,k+1} swz=0 ; rows {k+2,k+3} swz=16

  v8f acc[8];
#pragma unroll
  for (int tn = 0; tn < 8; ++tn)
#pragma unroll
    for (int r = 0; r < 8; ++r) acc[tn][r] = 0.0f;

  const float* __restrict__ Arow = A + (size_t)(tm * 16 + l15) * 128;

  for (int k = 0; k < 128; k += 4) {
    const int kk = k + half * 2;
    v2f a;
    a[0] = Arow[kk];
    a[1] = Arow[kk + 1];
#pragma unroll
    for (int tn = 0; tn < 8; ++tn) {
      const int c = (tn * 16 + l15) ^ swz;
      v2f b;
      b[0] = Bs[(kk << 7) | c];
      b[1] = Bs[((kk + 1) << 7) | c];
      acc[tn] = __builtin_amdgcn_wmma_f32_16x16x4_f32(false, a, false, b,
                                                      (short)0, acc[tn], false, false);
    }
  }

  // epilogue: h -> C, h*dinv^2 -> C2 (self-loop init of the aggregation buffer)
  float ds2[8];
#pragma unroll
  for (int r = 0; r < 8; ++r) {
    const float dv = dinv[tm * 16 + r + half * 8];
    ds2[r] = dv * dv;
  }
  float* __restrict__ Cp  = C  + (size_t)(tm * 16) * 128 + l15;
  float* __restrict__ C2p = C2 + (size_t)(tm * 16) * 128 + l15;
#pragma unroll
  for (int tn = 0; tn < 8; ++tn) {
#pragma unroll
    for (int r = 0; r < 8; ++r) {
      const size_t off = (size_t)(r + half * 8) * 128 + tn * 16;
      const float v = acc[tn][r];
      Cp[off]  = v;
      C2p[off] = v * ds2[r];
    }
  }
}

// ---------------- generic fp32 WMMA GEMM (head layer, K=256) ----------------
__global__ void gemm_wmma_f32(const float* __restrict__ A, const float* __restrict__ B,
                              const float* __restrict__ bias, float* __restrict__ C,
                              int M, int N, int K, int relu) {
  const int lane   = threadIdx.x & 31;
  const int wave   = threadIdx.x >> 5;
  const int tilesN = N >> 4;
  const int tilesM = M >> 4;
  const int t = blockIdx.x * (blockDim.x >> 5) + wave;
  if (t >= tilesM * tilesN) return;
  const int tm = t / tilesN;
  const int tn = t - tm * tilesN;

  const int half = lane >> 4;
  const int l15  = lane & 15;

  v8f acc;
  if (bias != nullptr) {
    const float bv = bias[tn * 16 + l15];
#pragma unroll
    for (int r = 0; r < 8; ++r) acc[r] = bv;
  } else {
#pragma unroll
    for (int r = 0; r < 8; ++r) acc[r] = 0.0f;
  }

  const float* __restrict__ Arow = A + (size_t)(tm * 16 + l15) * K;
  const float* __restrict__ Bcol = B + tn * 16 + l15;

  for (int k = 0; k < K; k += 4) {
    const int kk = k + half * 2;
    v2f a, b;
    a[0] = Arow[kk];
    a[1] = Arow[kk + 1];
    b[0] = Bcol[(size_t)kk * N];
    b[1] = Bcol[(size_t)(kk + 1) * N];
    acc = __builtin_amdgcn_wmma_f32_16x16x4_f32(false, a, false, b,
                                                (short)0, acc, false, false);
  }

  if (relu) {
#pragma unroll
    for (int r = 0; r < 8; ++r) acc[r] = acc[r] > 0.0f ? acc[r] : 0.0f;
  }

  float* __restrict__ Cp = C + (size_t)(tm * 16) * N + tn * 16 + l15;
#pragma unroll
  for (int r = 0; r < 8; ++r) Cp[(size_t)(r + half * 8) * N] = acc[r];
}

// ---------------- GCN aggregation ----------------

// one wave per edge; each lane moves float4 (32 lanes * 4 = 128 features)
__global__ void k_edge_scatter(const int* __restrict__ src, const int* __restrict__ dst,
                               const float* __restrict__ dinv, const float* __restrict__ h,
                               float* __restrict__ agg, int E) {
  const long tid = (long)blockIdx.x * blockDim.x + threadIdx.x;
  const int e    = (int)(tid >> 5);
  const int lane = (int)(tid & 31);
  if (e >= E) return;
  const int s = src[e];
  const int d = dst[e];
  const float w = dinv[s] * dinv[d];                // norm recomputed (L2-resident)
  const float4 v = ((const float4*)(h + (size_t)s * HID))[lane];
  float* ap = agg + (size_t)d * HID + lane * 4;
  atomicAdd(ap + 0, v.x * w);
  atomicAdd(ap + 1, v.y * w);
  atomicAdd(ap + 2, v.z * w);
  atomicAdd(ap + 3, v.w * w);
}

// in-place: agg = relu(agg + bias[f])
__global__ void k_finalize(float* __restrict__ agg, const float* __restrict__ bias, int total) {
  int i = blockIdx.x * blockDim.x + threadIdx.x;
  if (i >= total) return;
  const float v = agg[i] + bias[i & (HID - 1)];
  agg[i] = v > 0.0f ? v : 0.0f;
}

// ---------------- pooling ----------------

__global__ void k_count(const int* __restrict__ batch, float* __restrict__ counts, int N) {
  int i = blockIdx.x * blockDim.x + threadIdx.x;
  if (i < N) atomicAdd(&counts[batch[i]], 1.0f);
}

// h >= 0 post-ReLU, so int-ordered atomicMax on the bit pattern is a valid float max
__global__ void k_pool(const float* __restrict__ h, const int* __restrict__ batch,
                       float* __restrict__ sums, int* __restrict__ maxs, int total) {
  int i = blockIdx.x * blockDim.x + threadIdx.x;
  if (i >= total) return;
  const int node = i >> 7;
  const int f    = i & (HID - 1);
  const int g    = batch[node];
  const float v  = h[i];
  atomicAdd(&sums[g * HID + f], v);
  atomicMax(&maxs[g * HID + f], __float_as_int(v));
}

// gbuf[g, 0:128] = sums/max(counts,1); gbuf[g, 128:256] = max (0 for empty graphs)
__global__ void k_pool_final(const float* __restrict__ sums, const int* __restrict__ maxs,
                             const float* __restrict__ counts, float* __restrict__ gbuf,
                             int total /* NGR*HID */) {
  int i = blockIdx.x * blockDim.x + threadIdx.x;
  if (i >= total) return;
  const int g = i >> 7;
  const int f = i & (HID - 1);
  const float c = counts[g];
  const float denom = c > 1.0f ? c : 1.0f;
  gbuf[(size_t)g * (2 * HID) + f]       = sums[i] / denom;
  gbuf[(size_t)g * (2 * HID) + HID + f] = __int_as_float(maxs[i]);
}

// ---------------- classifier head, layer 2 (N_CLS=10 not tile-able) ----------------
__global__ void k_head(const float* __restrict__ hcls, const float* __restrict__ Wc2,
                       const float* __restrict__ bc2, float* __restrict__ out, int total) {
  int i = blockIdx.x * blockDim.x + threadIdx.x;
  if (i >= total) return;
  const int g = i / NCLS;
  const int c = i - g * NCLS;
  float s = bc2[c];
  const float* hv = hcls + (size_t)g * HID;
#pragma unroll 8
  for (int k = 0; k < HID; ++k) s += hv[k] * Wc2[k * NCLS + c];
  out[i] = s;
}

// ---------------- launch ----------------

extern "C" void kernel_launch(void* const* d_in, const int* in_sizes, int n_in,
                              void* d_out, int out_size, void* d_ws, size_t ws_size,
                              hipStream_t stream) {
  const float* x     = (const float*)d_in[0];
  const int*   edge  = (const int*)d_in[1];
  const int*   batch = (const int*)d_in[2];
  const float* W[3]  = {(const float*)d_in[3], (const float*)d_in[5], (const float*)d_in[7]};
  const float* b[3]  = {(const float*)d_in[4], (const float*)d_in[6], (const float*)d_in[8]};
  const float* Wc1   = (const float*)d_in[9];
  const float* bc1   = (const float*)d_in[10];
  const float* Wc2   = (const float*)d_in[11];
  const float* bc2   = (const float*)d_in[12];
  float* out = (float*)d_out;

  const int N = in_sizes[0] / HID;     // 100000 (multiple of 16)
  const int E = in_sizes[1] / 2;       // 1600000
  const int G = NGR;
  const int* src  = edge;
  const int* dstv = edge + E;

  // workspace carve-out (floats)
  float* ws     = (float*)d_ws;
  float* deg    = ws;
  float* dinv   = deg  + N;
  float* buf0   = dinv + N;                         // h = hin @ W
  float* buf1   = buf0 + (size_t)N * HID;           // agg / layer output (in place)
  float* counts = buf1 + (size_t)N * HID;
  float* sums   = counts + G;
  int*   maxs   = (int*)(sums + (size_t)G * HID);
  float* gbuf   = (float*)(maxs + (size_t)G * HID); // [G, 2*HID]
  float* hcls   = gbuf + (size_t)G * 2 * HID;       // [G, HID]

  const int BT = 256;
  auto nb = [](long n, int bt) { return (int)((n + bt - 1) / bt); };

  // degrees + normalization
  k_fill<<<nb(N, BT), BT, 0, stream>>>(deg, N, 0.0f);
  k_degree<<<nb(E, BT), BT, 0, stream>>>(dstv, deg, E);
  k_dinv<<<nb(N, BT), BT, 0, stream>>>(deg, dinv, N);

  // three GCN layers
  const int tot = N * HID;
  const long et = (long)E * 32;
  const int rowTiles = N / 16;                       // 6250
  const float* hin = x;
  for (int l = 0; l < 3; ++l) {
    gemm_wmma_strip_k128<<<(rowTiles + 7) / 8, 256, 0, stream>>>(hin, W[l],
                                                                 buf0, buf1, dinv, N);
    k_edge_scatter<<<nb(et, BT), BT, 0, stream>>>(src, dstv, dinv, buf0, buf1, E);
    k_finalize<<<nb(tot, BT), BT, 0, stream>>>(buf1, b[l], tot);
    hin = buf1;
  }

  // mean/max pooling over graphs
  k_fill<<<nb(G, BT), BT, 0, stream>>>(counts, G, 0.0f);
  k_fill<<<nb(G * HID, BT), BT, 0, stream>>>(sums, G * HID, 0.0f);
  k_fill<<<nb(G * HID, BT), BT, 0, stream>>>((float*)maxs, G * HID, 0.0f);
  k_count<<<nb(N, BT), BT, 0, stream>>>(batch, counts, N);
  k_pool<<<nb(tot, BT), BT, 0, stream>>>(buf1, batch, sums, maxs, tot);
  k_pool_final<<<nb(G * HID, BT), BT, 0, stream>>>(sums, maxs, counts, gbuf, G * HID);

  // classifier head
  const int tiles2 = (G / 16) * (HID / 16);
  gemm_wmma_f32<<<(tiles2 + 7) / 8, 256, 0, stream>>>(gbuf, Wc1, bc1, hcls,
                                                      G, HID, 2 * HID, 1);
  k_head<<<nb(G * NCLS, BT), BT, 0, stream>>>(hcls, Wc2, bc2, out, G * NCLS);
}